// ComputeSurfaceNormal_37331855737592
// MI455X (gfx1250) — compile-verified
//
#include <hip/hip_runtime.h>
#include <cstdint>

#define BATCH 32
#define HH    320
#define WW    1024
#define TY    10
#define NROWS (TY + 2)

typedef __attribute__((ext_vector_type(4))) float v4f;

// Async global->LDS copy (CDNA5 gfx1250 path, tracked by ASYNCcnt).
__device__ __forceinline__ void async_copy_f4_to_lds(uint32_t lds_byte_off,
                                                     const float* gptr) {
    asm volatile("global_load_async_to_lds_b128 %0, %1, off"
                 :: "v"(lds_byte_off), "v"((unsigned long long)(uintptr_t)gptr)
                 : "memory");
}
__device__ __forceinline__ void wait_asynccnt0() {
    asm volatile("s_wait_asynccnt 0" ::: "memory");
}

__global__ __launch_bounds__(256) void surfnorm_kernel(
    const float* __restrict__ depth,    // (B,1,H,W) raw in [-1,1]
    const float* __restrict__ invcamK,  // (B,4,4)
    float* __restrict__ out)            // (B,3,H,W)
{
    __shared__ __align__(16) float tile[NROWS * WW];   // 48 KB raw depth tile

    const int t   = threadIdx.x;       // 0..255
    const int b   = blockIdx.y;        // batch
    const int y0  = blockIdx.x * TY;   // first output row of this block
    const int col = t * 4;             // this thread owns columns col..col+3

    // ---------------- stage tile: rows y0-1 .. y0+TY, full width ----------------
    {
        const float* gbase = depth + (size_t)b * HH * WW + col;
        #pragma unroll
        for (int r = 0; r < NROWS; ++r) {
            const int gy = y0 - 1 + r;
            float* lptr = &tile[r * WW + col];
            if (gy >= 0 && gy < HH) {
                async_copy_f4_to_lds((uint32_t)(uintptr_t)lptr,
                                     gbase + (size_t)gy * WW);
            } else {
                *reinterpret_cast<v4f*>(lptr) = (v4f)(0.f);
            }
        }
    }
    wait_asynccnt0();
    __syncthreads();

    // ---------------- per-batch constants: 6 cross-product vectors ----------------
    const float* Km = invcamK + b * 16;  // row-major 4x4
    const float a0 = Km[0], b0 = Km[1], c0 = Km[2],  m0 = Km[3];
    const float a1 = Km[4], b1 = Km[5], c1 = Km[6],  m1 = Km[7];
    const float a2 = Km[8], b2 = Km[9], c2 = Km[10], m2 = Km[11];
    const float V1[3] = {a1*b2 - a2*b1, a2*b0 - a0*b2, a0*b1 - a1*b0}; // A x B
    const float V2[3] = {c1*a2 - c2*a1, c2*a0 - c0*a2, c0*a1 - c1*a0}; // C x A
    const float V3[3] = {c1*b2 - c2*b1, c2*b0 - c0*b2, c0*b1 - c1*b0}; // C x B
    const float V4[3] = {m1*a2 - m2*a1, m2*a0 - m0*a2, m0*a1 - m1*a0}; // M3 x A
    const float V5[3] = {m1*b2 - m2*b1, m2*b0 - m0*b2, m0*b1 - m1*b0}; // M3 x B
    const float V6[3] = {m1*c2 - m2*c1, m2*c0 - m0*c2, m0*c1 - m1*c0}; // M3 x C

    float* out0 = out + (size_t)(b * 3 + 0) * HH * WW;
    float* out1 = out + (size_t)(b * 3 + 1) * HH * WW;
    float* out2 = out + (size_t)(b * 3 + 2) * HH * WW;

    const float SC = 49.95f;   // depth scale
    const float OF = 50.05f;   // depth offset (handled via presence responses)

    // ----- per-column SAME-padding mask factors (only block-edge threads differ) -----
    const float pxm0 = (col > 0)          ? 1.f : 0.f;   // left halo valid  (t==0 edge)
    const float pxp3 = (col + 3 < WW - 1) ? 1.f : 0.f;   // right halo valid (t==255 edge)
    const int   xm   = (col > 0)          ? col - 1 : col;      // clamped, masked
    const int   xp   = (col + 4 <= WW - 1) ? col + 4 : WW - 1;  // clamped, masked
    const float Cd[4] = {1.f - pxm0, 0.f, 0.f, pxp3 - 1.f};  // (-1,0,1) col factor on mask
    const float Cs[4] = {1.f + pxm0, 2.f, 2.f, pxp3 + 1.f};  // ( 1,0,1) col factor on mask
    const float Cf[4] = {Cs[0] + 2.f, 4.f, 4.f, Cs[3] + 2.f};// ( 1,2,1) col factor on mask

    // rolling 3-row window of per-row combos (D=d+ - d-, S=d+ + d-, F=S+2*d0), 4 cols
    float D[3][4], S[3][4], F[3][4];

    #define LOADROW(RIDX, SLOT)                                                   \
        {                                                                         \
            const int _r = (RIDX);                                                \
            const v4f   v  = *reinterpret_cast<const v4f*>(&tile[_r*WW+col]);     \
            const float vL = tile[_r * WW + xm] * pxm0;                           \
            const float vR = tile[_r * WW + xp] * pxp3;                           \
            const float e0 = vL,  e1 = v.x, e2 = v.y,                             \
                        e3 = v.z, e4 = v.w, e5 = vR;                              \
            D[SLOT][0] = e2 - e0; S[SLOT][0] = e2 + e0;                           \
            D[SLOT][1] = e3 - e1; S[SLOT][1] = e3 + e1;                           \
            D[SLOT][2] = e4 - e2; S[SLOT][2] = e4 + e2;                           \
            D[SLOT][3] = e5 - e3; S[SLOT][3] = e5 + e3;                           \
            F[SLOT][0] = S[SLOT][0] + 2.f * e1;                                   \
            F[SLOT][1] = S[SLOT][1] + 2.f * e2;                                   \
            F[SLOT][2] = S[SLOT][2] + 2.f * e3;                                   \
            F[SLOT][3] = S[SLOT][3] + 2.f * e4;                                   \
        }

    LOADROW(0, 0)
    LOADROW(1, 1)

    #pragma unroll
    for (int rr = 0; rr < TY; ++rr) {
        const int iT = rr % 3;            // top    (dy = -1)
        const int iM = (rr + 1) % 3;      // middle (dy =  0)
        const int iB = (rr + 2) % 3;      // bottom (dy = +1)
        LOADROW(rr + 2, iB)

        const int   y   = y0 + rr;
        const float pym = (y > 0)      ? 1.f : 0.f;
        const float pyp = (y < HH - 1) ? 1.f : 0.f;
        const float Rd = pyp - pym, Rs = pyp + pym, R121 = Rs + 2.f;
        const float fy = (float)y;

        v4f r0, r1, r2;  // output channels for the 4 columns

        #pragma unroll
        for (int j = 0; j < 4; ++j) {
            // presence-mask stencil responses (exact SAME-padding borders)
            const float Sxp_ =  R121 * Cd[j];
            const float Txp_ =  R121 * Cs[j];
            const float Uxp_ =  Rd   * Cd[j];
            const float Syp_ = -Rd   * Cf[j];
            const float Uyp_ = -Rs   * Cf[j];

            // raw-depth stencil responses
            const float Sxr = D[iT][j] + 2.f * D[iM][j] + D[iB][j];
            const float Txr = S[iT][j] + 2.f * S[iM][j] + S[iB][j];
            const float Uxr = D[iB][j] - D[iT][j];
            const float Syr = F[iT][j] - F[iB][j];
            const float Uyr = -(F[iT][j] + F[iB][j]);

            // combined responses on rescaled masked depth
            const float Sx = SC * Sxr + OF * Sxp_;
            const float Tx = SC * Txr + OF * Txp_;
            const float Ux = SC * Uxr + OF * Uxp_;
            const float Sy = SC * Syr + OF * Syp_;
            const float Ty = -Ux;                     // identity: Ty == -Ux
            const float Uy = SC * Uyr + OF * Uyp_;
            const float Ex = Sxp_, Ey = Syp_;         // m3-column presence terms

            const float al = Sx * Ty - Tx * Sy;
            const float be = Sx * Uy - Ux * Sy;
            const float ga = Tx * Uy - Ux * Ty;
            const float dl = Ex * Sy - Sx * Ey;
            const float ep = Ex * Ty - Tx * Ey;
            const float zt = Ex * Uy - Ux * Ey;

            const float fx = (float)(col + j);
            const float s1 = be * fx - al * fy + ga;
            const float s4 = dl * fx + ep;
            const float s5 = dl * fy + zt;

            const float ox = s1*V1[0] + al*V2[0] + be*V3[0] + s4*V4[0] + s5*V5[0] + dl*V6[0];
            const float oy = s1*V1[1] + al*V2[1] + be*V3[1] + s4*V4[1] + s5*V5[1] + dl*V6[1];
            const float oz = s1*V1[2] + al*V2[2] + be*V3[2] + s4*V4[2] + s5*V5[2] + dl*V6[2];

            const float n2  = ox * ox + oy * oy + oz * oz;
            // 1/max(sqrt(n2),1e-12) == rsqrt(max(n2,1e-24)) (monotone), v_rsq_f32
            const float inv = __builtin_amdgcn_rsqf(fmaxf(n2, 1e-24f));

            r0[j] = ox * inv;
            r1[j] = oy * inv;
            r2[j] = oz * inv;
        }

        const size_t o = (size_t)y * WW + col;
        __builtin_nontemporal_store(r0, reinterpret_cast<v4f*>(out0 + o));
        __builtin_nontemporal_store(r1, reinterpret_cast<v4f*>(out1 + o));
        __builtin_nontemporal_store(r2, reinterpret_cast<v4f*>(out2 + o));
    }
    #undef LOADROW
}

extern "C" void kernel_launch(void* const* d_in, const int* in_sizes, int n_in,
                              void* d_out, int out_size, void* d_ws, size_t ws_size,
                              hipStream_t stream) {
    const float* depth = (const float*)d_in[0];   // (32,1,320,1024) fp32
    const float* invK  = (const float*)d_in[1];   // (32,4,4) fp32
    float* out = (float*)d_out;                   // (32,3,320,1024) fp32
    dim3 grid(HH / TY, BATCH);                    // 32 x 32 blocks
    surfnorm_kernel<<<grid, dim3(256), 0, stream>>>(depth, invK, out);
}